// QGNN_HetNodeClassifier_26740466385557
// MI455X (gfx1250) — compile-verified
//
#include <hip/hip_runtime.h>
#include <math.h>

#define N_UE 8192
#define N_AP 2048
#define KNBR 3
#define ETOT (N_UE*KNBR)

typedef __attribute__((ext_vector_type(2))) float v2f;
typedef __attribute__((ext_vector_type(8))) float v8f;

__device__ __forceinline__ float lrelu(float x){ return x > 0.0f ? x : 0.01f*x; }

__device__ __forceinline__ float2 cmul(float2 a, float2 b){
    return make_float2(fmaf(a.x, b.x, -a.y*b.y), fmaf(a.x, b.y, a.y*b.x));
}
__device__ __forceinline__ float2 cadd(float2 a, float2 b){ return make_float2(a.x+b.x, a.y+b.y); }

struct Gate { float2 m00, m01, m10, m11; };

__device__ __forceinline__ Gate g_rx(float t){
    float c = cosf(0.5f*t), s = sinf(0.5f*t);
    Gate g; g.m00 = make_float2(c,0.f); g.m01 = make_float2(0.f,-s);
            g.m10 = make_float2(0.f,-s); g.m11 = make_float2(c,0.f); return g;
}
__device__ __forceinline__ Gate g_ry(float t){
    float c = cosf(0.5f*t), s = sinf(0.5f*t);
    Gate g; g.m00 = make_float2(c,0.f); g.m01 = make_float2(-s,0.f);
            g.m10 = make_float2(s,0.f); g.m11 = make_float2(c,0.f); return g;
}
__device__ __forceinline__ Gate g_rz(float t){
    float c = cosf(0.5f*t), s = sinf(0.5f*t);
    Gate g; g.m00 = make_float2(c,-s); g.m01 = make_float2(0.f,0.f);
            g.m10 = make_float2(0.f,0.f); g.m11 = make_float2(c,s); return g;
}
// RZ(om) @ RY(th) @ RZ(phi)
__device__ __forceinline__ Gate g_rot(float phi, float th, float om){
    float c = cosf(0.5f*th), s = sinf(0.5f*th);
    float ap = 0.5f*(phi+om), am = 0.5f*(phi-om);
    float cap = cosf(ap), sap = sinf(ap), cam = cosf(am), sam = sinf(am);
    Gate g;
    g.m00 = make_float2( c*cap, -c*sap);   //  c * e^{-i(phi+om)/2}
    g.m01 = make_float2(-s*cam, -s*sam);   // -s * e^{ i(phi-om)/2}
    g.m10 = make_float2( s*cam, -s*sam);   //  s * e^{-i(phi-om)/2}
    g.m11 = make_float2( c*cap,  c*sap);   //  c * e^{ i(phi+om)/2}
    return g;
}

// qubit q lives at bit position (8-q); stride = 1<<bq. 256 pairs split 8/lane.
__device__ __forceinline__ void gate1(float2* S, int lane, int bq, Gate g){
    int stride = 1 << bq;
    #pragma unroll
    for (int t = 0; t < 8; ++t){
        int p  = (lane << 3) + t;
        int i0 = ((p >> bq) << (bq+1)) | (p & (stride-1));
        int i1 = i0 | stride;
        float2 a = S[i0], b = S[i1];
        S[i0] = cadd(cmul(g.m00,a), cmul(g.m01,b));
        S[i1] = cadd(cmul(g.m10,a), cmul(g.m11,b));
    }
}
__device__ __forceinline__ void gateC(float2* S, int lane, int bc, int bt, Gate g){
    int stride = 1 << bt;
    #pragma unroll
    for (int t = 0; t < 8; ++t){
        int p  = (lane << 3) + t;
        int i0 = ((p >> bt) << (bt+1)) | (p & (stride-1));
        if (!((i0 >> bc) & 1)) continue;
        int i1 = i0 | stride;
        float2 a = S[i0], b = S[i1];
        S[i0] = cadd(cmul(g.m00,a), cmul(g.m01,b));
        S[i1] = cadd(cmul(g.m10,a), cmul(g.m11,b));
    }
}
__device__ __forceinline__ void gateCX(float2* S, int lane, int bc, int bt){
    int stride = 1 << bt;
    #pragma unroll
    for (int t = 0; t < 8; ++t){
        int p  = (lane << 3) + t;
        int i0 = ((p >> bt) << (bt+1)) | (p & (stride-1));
        if (!((i0 >> bc) & 1)) continue;
        int i1 = i0 | stride;
        float2 a = S[i0]; S[i0] = S[i1]; S[i1] = a;
    }
}

__device__ __forceinline__ float wred(float v){
    #pragma unroll
    for (int m = 16; m > 0; m >>= 1) v += __shfl_xor(v, m, 32);
    return v;
}

// ------------------ Kernel 1: front-end node/edge MLPs ------------------
__global__ void __launch_bounds__(256) embed_kernel(
    const float* __restrict__ x_ue, const float* __restrict__ x_ap,
    const float* __restrict__ edge_attr,
    const float* __restrict__ Wn1u, const float* __restrict__ bn1u,
    const float* __restrict__ Wn2u, const float* __restrict__ bn2u,
    const float* __restrict__ Wn1a, const float* __restrict__ bn1a,
    const float* __restrict__ Wn2a, const float* __restrict__ bn2a,
    const float* __restrict__ We1,  const float* __restrict__ be1,
    const float* __restrict__ We2,  const float* __restrict__ be2,
    float* __restrict__ ue, float* __restrict__ ap, float* __restrict__ ea)
{
    int row = blockIdx.x * blockDim.x + threadIdx.x;
    const float *x, *W1, *b1, *W2, *b2; float* out; int D;
    if (row < N_UE) {
        x = x_ue + row*8; D = 8; W1=Wn1u; b1=bn1u; W2=Wn2u; b2=bn2u; out = ue + row*2;
    } else if (row < N_UE + N_AP) {
        int r = row - N_UE;
        x = x_ap + r*8; D = 8; W1=Wn1a; b1=bn1a; W2=Wn2a; b2=bn2a; out = ap + r*2;
    } else if (row < N_UE + N_AP + ETOT) {
        int r = row - N_UE - N_AP;
        x = edge_attr + r*4; D = 4; W1=We1; b1=be1; W2=We2; b2=be2; out = ea + r*2;
    } else return;

    float xr[8];
    #pragma unroll
    for (int k = 0; k < 8; ++k) xr[k] = (k < D) ? x[k] : 0.f;
    float o0 = 0.f, o1 = 0.f;
    for (int j = 0; j < 128; ++j){
        float h = b1[j];
        for (int k = 0; k < D; ++k) h = fmaf(xr[k], W1[k*128 + j], h);
        h = lrelu(h);
        o0 = fmaf(h, W2[2*j],   o0);
        o1 = fmaf(h, W2[2*j+1], o1);
    }
    out[0] = o0 + b2[0];
    out[1] = o1 + b2[1];
}

// ------------- Kernel 2: quantum circuit + update MLP + LayerNorm -------------
// One wave32 per UE node; 9-qubit statevector (512 x float2) in a private 4KB LDS slab.
__global__ void __launch_bounds__(256) circuit_kernel(
    const float* __restrict__ ue, const float* __restrict__ ap,
    const float* __restrict__ ea, const int* __restrict__ edge_src,
    const float* __restrict__ strong, const float* __restrict__ inits,
    const float* __restrict__ update,
    const float* __restrict__ Wu1, const float* __restrict__ bu1,
    const float* __restrict__ Wu2, const float* __restrict__ bu2,
    const float* __restrict__ ln_g, const float* __restrict__ ln_b,
    float* __restrict__ h_out)
{
    __shared__ float2 Sall[8 * 512];
    int wave = threadIdx.x >> 5, lane = threadIdx.x & 31;
    int node = blockIdx.x * 8 + wave;
    float2* S = Sall + wave * 512;

    // gather angles (uniform across the wave)
    float ef[3][2], nf[4][2];
    #pragma unroll
    for (int i = 0; i < 3; ++i){
        ef[i][0] = ea[(node*3+i)*2]; ef[i][1] = ea[(node*3+i)*2+1];
    }
    nf[0][0] = ue[node*2]; nf[0][1] = ue[node*2+1];
    #pragma unroll
    for (int i = 0; i < 3; ++i){
        int s = edge_src[node*3+i];
        nf[1+i][0] = ap[s*2]; nf[1+i][1] = ap[s*2+1];
    }

    // |0...0> init
    #pragma unroll
    for (int t = 0; t < 16; ++t) S[lane + 32*t] = make_float2(0.f, 0.f);
    if (lane == 0) S[0] = make_float2(1.f, 0.f);

    // feature encoding: edge qubits 0..2, node qubits 3..6 (bit = 8 - q)
    #pragma unroll
    for (int i = 0; i < 3; ++i){
        gate1(S, lane, 8-i, g_rx(ef[i][0]));
        gate1(S, lane, 8-i, g_rz(ef[i][1]));
    }
    #pragma unroll
    for (int i = 0; i < 4; ++i){
        gate1(S, lane, 5-i, g_rx(nf[i][0]));
        gate1(S, lane, 5-i, g_rz(nf[i][1]));
    }

    // per-edge entangling blocks
    Gate gcrx = g_rx(inits[0]), gcry = g_ry(inits[1]);
    #pragma unroll
    for (int i = 0; i < 3; ++i){
        int be = 8 - i;        // edge qubit i
        int bn = 4 - i;        // neighbor qubit 4+i
        gateC(S, lane, bn, be, gcrx);   // control=nb, target=e
        gateC(S, lane, be, bn, gcry);   // control=e,  target=nb
        for (int l = 0; l < 2; ++l){
            const float* w0 = strong + l*6;
            gate1(S, lane, be, g_rot(w0[0], w0[1], w0[2]));
            gate1(S, lane, bn, g_rot(w0[3], w0[4], w0[5]));
            gateCX(S, lane, be, bn);   // r = 1 for both layers (M=2)
            gateCX(S, lane, bn, be);
        }
    }

    // update blocks on wires [3, 4+i, 7, 8] -> bits {5, 4-i, 1, 0}
    #pragma unroll
    for (int i = 0; i < 3; ++i){
        int bw[4] = {5, 4 - i, 1, 0};
        for (int l = 0; l < 2; ++l){
            const float* w0 = update + i*24 + l*12;
            for (int j = 0; j < 4; ++j)
                gate1(S, lane, bw[j], g_rot(w0[j*3], w0[j*3+1], w0[j*3+2]));
            int r = l + 1;   // l%(M-1)+1 with M=4
            for (int j = 0; j < 4; ++j)
                gateCX(S, lane, bw[j], bw[(j + r) & 3]);
        }
    }

    // <Z> on qubits 3 (bit5), 7 (bit1), 8 (bit0)
    float a3 = 0.f, a7 = 0.f, a8 = 0.f;
    #pragma unroll
    for (int t = 0; t < 16; ++t){
        int idx = lane + 32*t;
        float2 v = S[idx];
        float p = v.x*v.x + v.y*v.y;
        a3 += ((idx >> 5) & 1) ? -p : p;
        a7 += ((idx >> 1) & 1) ? -p : p;
        a8 += ( idx       & 1) ? -p : p;
    }
    a3 = wred(a3); a7 = wred(a7); a8 = wred(a8);

    // fused update MLP (5 -> 128 -> 2), residual, LayerNorm over 2 dims
    float in5[5] = { nf[0][0], nf[0][1], a3, a7, a8 };
    float u0 = 0.f, u1 = 0.f;
    #pragma unroll
    for (int t = 0; t < 4; ++t){
        int j = lane + 32*t;
        float h = bu1[j];
        #pragma unroll
        for (int k = 0; k < 5; ++k) h = fmaf(in5[k], Wu1[k*128 + j], h);
        h = lrelu(h);
        u0 = fmaf(h, Wu2[2*j],   u0);
        u1 = fmaf(h, Wu2[2*j+1], u1);
    }
    u0 = wred(u0); u1 = wred(u1);
    float h0 = nf[0][0] + u0 + bu2[0];
    float h1 = nf[0][1] + u1 + bu2[1];
    float mu = 0.5f*(h0 + h1);
    float d0 = h0 - mu, d1 = h1 - mu;
    float rs = rsqrtf(0.5f*(d0*d0 + d1*d1) + 1e-5f);
    if (lane == 0){
        h_out[node*2]   = d0*rs*ln_g[0] + ln_b[0];
        h_out[node*2+1] = d1*rs*ln_g[1] + ln_b[1];
    }
}

// ------------- Kernel 3: head MLP 2->128->128->2, WMMA f32 16x16x4 -------------
// 4 waves/block, each wave owns a 16-row M-tile. z1/z2 staged in a private LDS slab.
__global__ void __launch_bounds__(128) head_kernel(
    const float* __restrict__ h_in,
    const float* __restrict__ Wf1, const float* __restrict__ bf1,
    const float* __restrict__ Wf2, const float* __restrict__ bf2,
    const float* __restrict__ Wf3, const float* __restrict__ bf3,
    float* __restrict__ out)
{
    __shared__ float zbuf[4][16*128];
    int wave = threadIdx.x >> 5, lane = threadIdx.x & 31;
    int row0 = blockIdx.x * 64 + wave * 16;
    float* Z = zbuf[wave];

    __builtin_prefetch(Wf2 + (threadIdx.x << 7), 0, 0);   // warm GL2 for the GEMM weights

    // layer 1: z1 = lrelu(h @ Wf1 + bf1), K=2 -> plain VALU into LDS
    #pragma unroll
    for (int t = 0; t < 4; ++t){
        int j = lane + 32*t;
        float w0 = Wf1[j], w1 = Wf1[128 + j], bb = bf1[j];
        for (int m = 0; m < 16; ++m){
            float h0 = h_in[(row0+m)*2], h1 = h_in[(row0+m)*2 + 1];
            Z[m*128 + j] = lrelu(fmaf(h0, w0, fmaf(h1, w1, bb)));
        }
    }

    // layer 2: z2 = lrelu(z1 @ Wf2 + bf2), 16x128x128 via V_WMMA_F32_16X16X4_F32
    int half = lane >> 4, mr = lane & 15;
    v8f acc[8];
    #pragma unroll
    for (int nt = 0; nt < 8; ++nt){
        v8f c = {0.f,0.f,0.f,0.f,0.f,0.f,0.f,0.f};
        int n = nt*16 + mr;
        for (int k = 0; k < 128; k += 4){
            int kk = k + 2*half;
            v2f a; a.x = Z[mr*128 + kk];       a.y = Z[mr*128 + kk + 1];
            v2f b; b.x = Wf2[kk*128 + n];      b.y = Wf2[(kk+1)*128 + n];
            c = __builtin_amdgcn_wmma_f32_16x16x4_f32(
                    false, a, false, b, (short)0, c, false, false);
        }
        float bb = bf2[n];
        #pragma unroll
        for (int r = 0; r < 8; ++r) c[r] = lrelu(c[r] + bb);
        acc[nt] = c;
    }

    // scatter z2 fragments back into the (now dead) z1 LDS slab
    #pragma unroll
    for (int nt = 0; nt < 8; ++nt)
        #pragma unroll
        for (int r = 0; r < 8; ++r)
            Z[(r + 8*half)*128 + nt*16 + mr] = acc[nt][r];

    // layer 3: out = sigmoid(z2 @ Wf3 + bf3); one (row, class) per lane
    int m = lane >> 1, cc = lane & 1;
    float s = bf3[cc];
    for (int k = 0; k < 128; ++k) s = fmaf(Z[m*128 + k], Wf3[k*2 + cc], s);
    out[(row0 + m)*2 + cc] = 1.f / (1.f + expf(-s));
}

extern "C" void kernel_launch(void* const* d_in, const int* in_sizes, int n_in,
                              void* d_out, int out_size, void* d_ws, size_t ws_size,
                              hipStream_t stream)
{
    (void)in_sizes; (void)n_in; (void)out_size; (void)ws_size;
    const float* x_ue   = (const float*)d_in[0];
    const float* x_ap   = (const float*)d_in[1];
    const float* e_attr = (const float*)d_in[2];
    const int*   e_src  = (const int*)  d_in[3];
    // d_in[4] = edge_dst (unused by the reference)
    const float* Wn1u = (const float*)d_in[5],  *bn1u = (const float*)d_in[6];
    const float* Wn2u = (const float*)d_in[7],  *bn2u = (const float*)d_in[8];
    const float* Wn1a = (const float*)d_in[9],  *bn1a = (const float*)d_in[10];
    const float* Wn2a = (const float*)d_in[11], *bn2a = (const float*)d_in[12];
    const float* We1  = (const float*)d_in[13], *be1  = (const float*)d_in[14];
    const float* We2  = (const float*)d_in[15], *be2  = (const float*)d_in[16];
    const float* strong = (const float*)d_in[17];
    const float* inits  = (const float*)d_in[18];
    const float* update = (const float*)d_in[19];
    const float* Wu1 = (const float*)d_in[20], *bu1 = (const float*)d_in[21];
    const float* Wu2 = (const float*)d_in[22], *bu2 = (const float*)d_in[23];
    const float* ln_g = (const float*)d_in[24], *ln_b = (const float*)d_in[25];
    const float* Wf1 = (const float*)d_in[26], *bf1 = (const float*)d_in[27];
    const float* Wf2 = (const float*)d_in[28], *bf2 = (const float*)d_in[29];
    const float* Wf3 = (const float*)d_in[30], *bf3 = (const float*)d_in[31];
    float* outp = (float*)d_out;

    float* ws  = (float*)d_ws;
    float* ue  = ws;                   // 8192*2
    float* ap  = ws + 16384;           // 2048*2
    float* ea  = ws + 20480;           // 24576*2
    float* hln = ws + 69632;           // 8192*2

    embed_kernel<<<(N_UE + N_AP + ETOT) / 256, 256, 0, stream>>>(
        x_ue, x_ap, e_attr,
        Wn1u, bn1u, Wn2u, bn2u, Wn1a, bn1a, Wn2a, bn2a, We1, be1, We2, be2,
        ue, ap, ea);

    circuit_kernel<<<N_UE / 8, 256, 0, stream>>>(
        ue, ap, ea, e_src, strong, inits, update,
        Wu1, bu1, Wu2, bu2, ln_g, ln_b, hln);

    head_kernel<<<N_UE / 64, 128, 0, stream>>>(
        hln, Wf1, bf1, Wf2, bf2, Wf3, bf3, outp);
}